// TransitionDown_72919954752184
// MI455X (gfx1250) — compile-verified
//
#include <hip/hip_runtime.h>
#include <hip/hip_bf16.h>
#include <cstdint>

typedef __attribute__((ext_vector_type(2))) float v2f;
typedef __attribute__((ext_vector_type(8))) float v8f;

#define B_      8
#define N_      8192
#define CIN     64
#define COUT    128
#define NPOINT  2048
#define K_      16
#define PPB     4        // centroids (16-row WMMA tiles) per block in GEMM kernels
#define BN_EPS_ 1e-5f

// ---------------------------------------------------------------------------
// Kernel 1: farthest point sampling. One block per batch, 1024 threads,
// 8 points per thread held entirely in registers. Per iteration:
// wave32 shuffle argmax -> 32 partials in LDS -> wave0 shuffle argmax.
// Writes new_xyz directly into d_out.
// ---------------------------------------------------------------------------
__global__ __launch_bounds__(1024) void fps_kernel(
    const float* __restrict__ xyz, float* __restrict__ new_xyz) {
  const int b   = blockIdx.x;
  const int tid = threadIdx.x;
  const int PT  = N_ / 1024;  // 8 points per thread
  const float* xb = xyz + (size_t)b * N_ * 3;

  float px[8], py[8], pz[8], dist[8];
#pragma unroll
  for (int i = 0; i < PT; ++i) {
    int p = tid * PT + i;
    px[i] = xb[p * 3 + 0];
    py[i] = xb[p * 3 + 1];
    pz[i] = xb[p * 3 + 2];
    dist[i] = 1e10f;
  }

  __shared__ float s_val[32];
  __shared__ int   s_idx[32];
  __shared__ float s_cent[3];
  __shared__ int   s_win;

  // deterministic seed point = index 0 (matches reference far0 = 0)
  float cx = xb[0], cy = xb[1], cz = xb[2];

  for (int it = 0; it < NPOINT; ++it) {
    if (tid == 0) {
      float* o = new_xyz + ((size_t)b * NPOINT + it) * 3;
      o[0] = cx; o[1] = cy; o[2] = cz;
    }
    // update running min-distance, track local argmax (first-max tie rule)
    float bestd = -1.0f;
    int   besti = 0;
#pragma unroll
    for (int i = 0; i < PT; ++i) {
      float dx = px[i] - cx, dy = py[i] - cy, dz = pz[i] - cz;
      float d  = dx * dx + dy * dy + dz * dz;
      dist[i]  = fminf(dist[i], d);
      if (dist[i] > bestd) { bestd = dist[i]; besti = tid * PT + i; }
    }
    // wave32 argmax
#pragma unroll
    for (int off = 16; off > 0; off >>= 1) {
      float od = __shfl_down(bestd, off, 32);
      int   oi = __shfl_down(besti, off, 32);
      if (od > bestd || (od == bestd && oi < besti)) { bestd = od; besti = oi; }
    }
    if ((tid & 31) == 0) { s_val[tid >> 5] = bestd; s_idx[tid >> 5] = besti; }
    __syncthreads();
    if (tid < 32) {
      bestd = s_val[tid];
      besti = s_idx[tid];
#pragma unroll
      for (int off = 16; off > 0; off >>= 1) {
        float od = __shfl_down(bestd, off, 32);
        int   oi = __shfl_down(besti, off, 32);
        if (od > bestd || (od == bestd && oi < besti)) { bestd = od; besti = oi; }
      }
      if (tid == 0) s_win = besti;
    }
    __syncthreads();
    int win = s_win;
    if ((win >> 3) == tid) {  // owner thread broadcasts winner coords
      float ox = 0.f, oy = 0.f, oz = 0.f;
#pragma unroll
      for (int i = 0; i < PT; ++i)
        if ((win & 7) == i) { ox = px[i]; oy = py[i]; oz = pz[i]; }
      s_cent[0] = ox; s_cent[1] = oy; s_cent[2] = oz;
    }
    __syncthreads();
    cx = s_cent[0]; cy = s_cent[1]; cz = s_cent[2];
  }
}

// ---------------------------------------------------------------------------
// Kernel 2: KNN, one thread per centroid, 256-point LDS tiles,
// fully-unrolled 16-entry register insertion sort (constant indices only).
// ---------------------------------------------------------------------------
__global__ __launch_bounds__(256) void knn_kernel(
    const float* __restrict__ xyz, const float* __restrict__ new_xyz,
    int* __restrict__ knn_idx) {
  const int gid = blockIdx.x * 256 + threadIdx.x;  // [0, B*NPOINT)
  const int b   = blockIdx.x / (NPOINT / 256);     // 8 blocks per batch
  const float qx = new_xyz[(size_t)gid * 3 + 0];
  const float qy = new_xyz[(size_t)gid * 3 + 1];
  const float qz = new_xyz[(size_t)gid * 3 + 2];
  const float* xb = xyz + (size_t)b * N_ * 3;

  float bd[K_];
  int   bi[K_];
#pragma unroll
  for (int j = 0; j < K_; ++j) { bd[j] = 1e30f; bi[j] = 0; }

  __shared__ float sx[256], sy[256], sz[256];
  for (int t0 = 0; t0 < N_; t0 += 256) {
    __syncthreads();
    int p = t0 + threadIdx.x;
    sx[threadIdx.x] = xb[p * 3 + 0];
    sy[threadIdx.x] = xb[p * 3 + 1];
    sz[threadIdx.x] = xb[p * 3 + 2];
    __syncthreads();
    for (int j = 0; j < 256; ++j) {
      float dx = sx[j] - qx, dy = sy[j] - qy, dz = sz[j] - qz;
      float d  = dx * dx + dy * dy + dz * dz;
      if (d < bd[K_ - 1]) {
        bd[K_ - 1] = d;
        bi[K_ - 1] = t0 + j;
#pragma unroll
        for (int u = K_ - 1; u > 0; --u) {
          if (bd[u] < bd[u - 1]) {
            float td = bd[u]; bd[u] = bd[u - 1]; bd[u - 1] = td;
            int   ti = bi[u]; bi[u] = bi[u - 1]; bi[u - 1] = ti;
          }
        }
      }
    }
  }
#pragma unroll
  for (int j = 0; j < K_; ++j) knn_idx[(size_t)gid * K_ + j] = bi[j];
}

// ---------------------------------------------------------------------------
// Kernel 3: zero BN stat accumulators (d_ws is poisoned, must init).
// ---------------------------------------------------------------------------
__global__ void zero_stats_kernel(float* __restrict__ stats) {
  if (threadIdx.x < 2 * COUT) stats[threadIdx.x] = 0.0f;
}

// ---------------------------------------------------------------------------
// Shared staging for the GEMM kernels: W (64x128) and PPB gathered A-tiles
// (each 16 neighbors x 64 channels) into LDS.  48 KB total.
// ---------------------------------------------------------------------------
__device__ __forceinline__ void stage_tiles(
    const float* __restrict__ feature, const float* __restrict__ W,
    const int* __restrict__ knn_idx, int gpt0, int b, int tid,
    float* sW, float* sA, int* sIdx) {
  __builtin_prefetch(W + tid, 0, 0);  // global_prefetch_b8: warm W stream
  for (int e = tid; e < CIN * COUT; e += 256) sW[e] = W[e];
  if (tid < PPB * K_) sIdx[tid] = knn_idx[(size_t)gpt0 * K_ + tid];
  __syncthreads();
  for (int e = tid; e < PPB * K_ * CIN; e += 256) {
    int row = e >> 6, c = e & 63;  // row in [0, PPB*K_), c in [0, CIN)
    sA[e] = feature[((size_t)b * N_ + sIdx[row]) * CIN + c];
  }
  __syncthreads();
}

// ---------------------------------------------------------------------------
// Kernel 4: fused gather + f32 WMMA GEMM (+bias) -> per-channel sum/sumsq.
// One block per PPB centroids; 8 waves each own one 16-col tile of Cout=128
// and sweep the PPB 16-row tiles reusing their B fragments.
// ---------------------------------------------------------------------------
__global__ __launch_bounds__(256) void gemm_stats_kernel(
    const float* __restrict__ feature, const float* __restrict__ W,
    const float* __restrict__ bias, const int* __restrict__ knn_idx,
    float* __restrict__ sums, float* __restrict__ sumsq) {
  __shared__ float sW[CIN * COUT];      // 32 KB
  __shared__ float sA[PPB * K_ * CIN];  // 16 KB
  __shared__ int   sIdx[PPB * K_];
  const int tid  = threadIdx.x;
  const int gpt0 = blockIdx.x * PPB;    // first centroid of this block
  const int b    = gpt0 / NPOINT;       // block never straddles batches

  stage_tiles(feature, W, knn_idx, gpt0, b, tid, sW, sA, sIdx);

  const int wave = tid >> 5, lane = tid & 31;
  const int half = lane >> 4, r = lane & 15;
  const int col  = wave * 16 + r;
  const int klo  = 2 * half;

  // load the 16 B fragments once; reuse across PPB tile sweeps
  v2f wf[CIN / 4];
#pragma unroll
  for (int kb = 0; kb < CIN / 4; ++kb) {
    wf[kb].x = sW[(kb * 4 + klo) * COUT + col];
    wf[kb].y = sW[(kb * 4 + klo + 1) * COUT + col];
  }

  const float bb = bias[col];
  float s = 0.f, q = 0.f;
#pragma unroll
  for (int p = 0; p < PPB; ++p) {
    const float* A = &sA[p * K_ * CIN];
    v8f acc;
#pragma unroll
    for (int i = 0; i < 8; ++i) acc[i] = bb;
#pragma unroll
    for (int kb = 0; kb < CIN / 4; ++kb) {
      v2f a;
      a.x = A[r * CIN + kb * 4 + klo];
      a.y = A[r * CIN + kb * 4 + klo + 1];
      acc = __builtin_amdgcn_wmma_f32_16x16x4_f32(
          false, a, false, wf[kb], (short)0, acc, false, false);
    }
#pragma unroll
    for (int i = 0; i < 8; ++i) { s += acc[i]; q += acc[i] * acc[i]; }
  }
  s += __shfl_xor(s, 16, 32);  // combine M=0..7 with M=8..15 (same column)
  q += __shfl_xor(q, 16, 32);
  if (half == 0) {
    atomicAdd(&sums[col], s);
    atomicAdd(&sumsq[col], q);
  }
}

// ---------------------------------------------------------------------------
// Kernel 5: finalize BN stats into per-channel scale/shift.
// ---------------------------------------------------------------------------
__global__ void finalize_stats_kernel(
    const float* __restrict__ sums, const float* __restrict__ sumsq,
    const float* __restrict__ gamma, const float* __restrict__ beta,
    float* __restrict__ scl, float* __restrict__ shf) {
  int c = threadIdx.x;
  if (c < COUT) {
    const float cnt  = (float)((size_t)B_ * NPOINT * K_);
    float mean = sums[c] / cnt;
    float var  = sumsq[c] / cnt - mean * mean;
    float iv   = gamma[c] * rsqrtf(var + BN_EPS_);
    scl[c] = iv;
    shf[c] = beta[c] - mean * iv;
  }
}

// ---------------------------------------------------------------------------
// Kernel 6: recompute GEMM, apply BN + ReLU, max over the 16 neighbors
// (each tile's M dimension), write new_feature.
// ---------------------------------------------------------------------------
__global__ __launch_bounds__(256) void gemm_out_kernel(
    const float* __restrict__ feature, const float* __restrict__ W,
    const float* __restrict__ bias, const int* __restrict__ knn_idx,
    const float* __restrict__ scl, const float* __restrict__ shf,
    float* __restrict__ new_feat) {
  __shared__ float sW[CIN * COUT];
  __shared__ float sA[PPB * K_ * CIN];
  __shared__ int   sIdx[PPB * K_];
  const int tid  = threadIdx.x;
  const int gpt0 = blockIdx.x * PPB;
  const int b    = gpt0 / NPOINT;

  stage_tiles(feature, W, knn_idx, gpt0, b, tid, sW, sA, sIdx);

  const int wave = tid >> 5, lane = tid & 31;
  const int half = lane >> 4, r = lane & 15;
  const int col  = wave * 16 + r;
  const int klo  = 2 * half;

  v2f wf[CIN / 4];
#pragma unroll
  for (int kb = 0; kb < CIN / 4; ++kb) {
    wf[kb].x = sW[(kb * 4 + klo) * COUT + col];
    wf[kb].y = sW[(kb * 4 + klo + 1) * COUT + col];
  }

  const float bb = bias[col];
  const float a_ = scl[col], sh = shf[col];
#pragma unroll
  for (int p = 0; p < PPB; ++p) {
    const float* A = &sA[p * K_ * CIN];
    v8f acc;
#pragma unroll
    for (int i = 0; i < 8; ++i) acc[i] = bb;
#pragma unroll
    for (int kb = 0; kb < CIN / 4; ++kb) {
      v2f a;
      a.x = A[r * CIN + kb * 4 + klo];
      a.y = A[r * CIN + kb * 4 + klo + 1];
      acc = __builtin_amdgcn_wmma_f32_16x16x4_f32(
          false, a, false, wf[kb], (short)0, acc, false, false);
    }
    float mx = 0.0f;  // ReLU outputs are >= 0
#pragma unroll
    for (int i = 0; i < 8; ++i) {
      float y = fmaxf(acc[i] * a_ + sh, 0.0f);
      mx = fmaxf(mx, y);
    }
    mx = fmaxf(mx, __shfl_xor(mx, 16, 32));  // max over all 16 neighbors
    if (half == 0) new_feat[(size_t)(gpt0 + p) * COUT + col] = mx;
  }
}

// ---------------------------------------------------------------------------
extern "C" void kernel_launch(void* const* d_in, const int* in_sizes, int n_in,
                              void* d_out, int out_size, void* d_ws, size_t ws_size,
                              hipStream_t stream) {
  const float* xyz     = (const float*)d_in[0];
  const float* feature = (const float*)d_in[1];
  const float* W       = (const float*)d_in[2];
  const float* bias    = (const float*)d_in[3];
  const float* gamma   = (const float*)d_in[4];
  const float* beta    = (const float*)d_in[5];
  (void)in_sizes; (void)n_in; (void)out_size; (void)ws_size;

  float* out      = (float*)d_out;
  float* new_xyz  = out;                              // [B, NPOINT, 3]
  float* new_feat = out + (size_t)B_ * NPOINT * 3;    // [B, NPOINT, COUT]

  int*   knn_idx = (int*)d_ws;                               // [B, NPOINT, K]
  float* stats   = (float*)((char*)d_ws +
                            (size_t)B_ * NPOINT * K_ * sizeof(int));
  float* sums  = stats;
  float* sumsq = stats + COUT;
  float* scl   = stats + 2 * COUT;
  float* shf   = stats + 3 * COUT;

  fps_kernel<<<B_, 1024, 0, stream>>>(xyz, new_xyz);
  knn_kernel<<<(B_ * NPOINT) / 256, 256, 0, stream>>>(xyz, new_xyz, knn_idx);
  zero_stats_kernel<<<1, 256, 0, stream>>>(sums);
  gemm_stats_kernel<<<(B_ * NPOINT) / PPB, 256, 0, stream>>>(
      feature, W, bias, knn_idx, sums, sumsq);
  finalize_stats_kernel<<<1, 128, 0, stream>>>(sums, sumsq, gamma, beta, scl, shf);
  gemm_out_kernel<<<(B_ * NPOINT) / PPB, 256, 0, stream>>>(
      feature, W, bias, knn_idx, scl, shf, new_feat);
}